// Network_21010980012235
// MI455X (gfx1250) — compile-verified
//
#include <hip/hip_runtime.h>
#include <hip/hip_bf16.h>
#include <cstdint>
#include <cfloat>

#define NUM_CLUSTERS 256
#define TILE_ROWS 32
#define ROW_W 36            // 35 output floats + 1 pad (bank spread)
#define WAVES_PER_BLOCK 8

// ---------------------------------------------------------------------------
// lower_bound over the sorted cluster-id column of clusters_idx (int2: {cid,pid})
// ---------------------------------------------------------------------------
__device__ __forceinline__ int lower_bound_cid(const int2* __restrict__ cidx,
                                               int n, int key) {
  int lo = 0, hi = n;
  while (lo < hi) {
    int mid = (lo + hi) >> 1;
    if (cidx[mid].x < key) lo = mid + 1; else hi = mid;
  }
  return lo;
}

// ---------------------------------------------------------------------------
// Per-cluster stats: one block per cluster (cid sorted -> contiguous range).
// Deterministic tree reductions (no atomics). Writes per-cluster:
//   mean4[c] = {mean.x, mean.y, mean.z, 0}
//   sct4[c]  = {scale, off.x, off.y, off.z}   (off = -coords_min*scale)
// ---------------------------------------------------------------------------
__global__ __launch_bounds__(256) void stats_kernel(
    const int2* __restrict__ cidx, const float* __restrict__ coords,
    const int* __restrict__ fullscale_p, const int* __restrict__ scale_p,
    float* __restrict__ mean4, float* __restrict__ sct4, int n) {
  __shared__ float4 red[256];
  const int c = blockIdx.x;
  const int tid = threadIdx.x;

  const int lo = lower_bound_cid(cidx, n, c);
  const int hi = lower_bound_cid(cidx, n, c + 1);
  const int cnt = hi - lo;

  // ---- pass 1: sum of coords ----
  float sx = 0.f, sy = 0.f, sz = 0.f;
  for (int i = lo + tid; i < hi; i += 256) {
    const int pid = cidx[i].y;
    sx += coords[pid * 3 + 0];
    sy += coords[pid * 3 + 1];
    sz += coords[pid * 3 + 2];
  }
  red[tid] = make_float4(sx, sy, sz, 0.f);
  __syncthreads();
  for (int off = 128; off > 0; off >>= 1) {
    if (tid < off) {
      float4 a = red[tid], b = red[tid + off];
      red[tid] = make_float4(a.x + b.x, a.y + b.y, a.z + b.z, 0.f);
    }
    __syncthreads();
  }
  const float inv = (cnt > 0) ? (1.0f / (float)cnt) : 0.0f;
  const float cmx = red[0].x * inv;
  const float cmy = red[0].y * inv;
  const float cmz = red[0].z * inv;
  __syncthreads();

  // ---- pass 2: min/max of centered coords ----
  float mnx = FLT_MAX, mny = FLT_MAX, mnz = FLT_MAX;
  float mxx = -FLT_MAX, mxy = -FLT_MAX, mxz = -FLT_MAX;
  for (int i = lo + tid; i < hi; i += 256) {
    const int pid = cidx[i].y;
    const float x = coords[pid * 3 + 0] - cmx;
    const float y = coords[pid * 3 + 1] - cmy;
    const float z = coords[pid * 3 + 2] - cmz;
    mnx = fminf(mnx, x); mny = fminf(mny, y); mnz = fminf(mnz, z);
    mxx = fmaxf(mxx, x); mxy = fmaxf(mxy, y); mxz = fmaxf(mxz, z);
  }
  red[tid] = make_float4(mnx, mny, mnz, 0.f);
  __syncthreads();
  for (int off = 128; off > 0; off >>= 1) {
    if (tid < off) {
      float4 a = red[tid], b = red[tid + off];
      red[tid] = make_float4(fminf(a.x, b.x), fminf(a.y, b.y), fminf(a.z, b.z), 0.f);
    }
    __syncthreads();
  }
  const float4 mn = red[0];
  __syncthreads();
  red[tid] = make_float4(mxx, mxy, mxz, 0.f);
  __syncthreads();
  for (int off = 128; off > 0; off >>= 1) {
    if (tid < off) {
      float4 a = red[tid], b = red[tid + off];
      red[tid] = make_float4(fmaxf(a.x, b.x), fmaxf(a.y, b.y), fmaxf(a.z, b.z), 0.f);
    }
    __syncthreads();
  }

  if (tid == 0) {
    const float4 mx = red[0];
    mean4[c * 4 + 0] = cmx;
    mean4[c * 4 + 1] = cmy;
    mean4[c * 4 + 2] = cmz;
    mean4[c * 4 + 3] = 0.f;
    const float fs = (float)fullscale_p[0];
    // max((max-min)/fs) == max(max-min)/fs exactly (monotone positive division)
    const float d = fmaxf(fmaxf(mx.x - mn.x, mx.y - mn.y), mx.z - mn.z) / fs;
    float s = 1.0f / d - 0.01f;         // d==0 -> +inf, clamped below (matches jnp)
    s = fminf(s, (float)scale_p[0]);
    sct4[c * 4 + 0] = s;
    sct4[c * 4 + 1] = -mn.x * s;
    sct4[c * 4 + 2] = -mn.y * s;
    sct4[c * 4 + 3] = -mn.z * s;
  }
}

// ---------------------------------------------------------------------------
// Output kernel: one wave per 32-row tile.
//   * feats gather: 32 async global->LDS copies (128B coalesced each) — CDNA5
//     ASYNCcnt path (global_load_async_to_lds_b32 / s_wait_asynccnt). Default
//     RT caching: the 128MB feats table stays resident in the 192MB L2.
//   * coords transform written into same LDS tile while gather is in flight.
//   * 32x35 tile streamed out with coalesced NON-TEMPORAL stores so the
//     210MB write-once output does not evict the feats table from L2.
// ---------------------------------------------------------------------------
__global__ __launch_bounds__(256) void out_kernel(
    const int2* __restrict__ cidx, const float* __restrict__ feats,
    const float* __restrict__ coords, const float4* __restrict__ mean4,
    const float4* __restrict__ sct4, float* __restrict__ out, int nrows) {
  __shared__ float srow[WAVES_PER_BLOCK][TILE_ROWS * ROW_W];  // 36864 B/block
  const int lane = threadIdx.x & 31;
  const int w = threadIdx.x >> 5;
  const int nTiles = nrows >> 5;
  const unsigned ldsBase = (unsigned)(uintptr_t)&srow[w][0];  // LDS byte offset

  for (int tile = blockIdx.x * WAVES_PER_BLOCK + w; tile < nTiles;
       tile += gridDim.x * WAVES_PER_BLOCK) {
    const int row = (tile << 5) + lane;
    // idx is streamed once -> non-temporal load
    const long long raw =
        __builtin_nontemporal_load((const long long*)&cidx[row]);
    const int cid = (int)(unsigned)(raw & 0xffffffffll);
    const int pid = (int)(unsigned)((unsigned long long)raw >> 32);

    // 32 async gathers: tile-row j <- feats[pid_j][0:32), lane-per-float.
#pragma unroll
    for (int j = 0; j < TILE_ROWS; ++j) {
      const int pj = __shfl(pid, j);
      const unsigned long long src =
          (unsigned long long)(feats + ((size_t)(unsigned)pj << 5) + (unsigned)lane);
      const unsigned dst = ldsBase + (unsigned)((j * ROW_W + 3 + lane) << 2);
      asm volatile("global_load_async_to_lds_b32 %0, %1, off"
                   :: "v"(dst), "v"(src)
                   : "memory");
    }

    // coords transform (reference order: (c - mean)*scale + offset) while
    // the async gather is in flight.
    const float4 m = mean4[cid];
    const float4 s = sct4[cid];
    const float cx = coords[pid * 3 + 0];
    const float cy = coords[pid * 3 + 1];
    const float cz = coords[pid * 3 + 2];
    srow[w][lane * ROW_W + 0] = (cx - m.x) * s.x + s.y;
    srow[w][lane * ROW_W + 1] = (cy - m.y) * s.x + s.z;
    srow[w][lane * ROW_W + 2] = (cz - m.z) * s.x + s.w;

    asm volatile("s_wait_asynccnt 0" ::: "memory");

    // Coalesced non-temporal store of the 32x35 tile (1120 contiguous dwords).
    const size_t ob = (size_t)(tile << 5) * 35u;
#pragma unroll
    for (int i = 0; i < 35; ++i) {
      const int t = i * 32 + lane;
      const int rr = t / 35;
      const int cc = t - rr * 35;
      __builtin_nontemporal_store(srow[w][rr * ROW_W + cc], &out[ob + (unsigned)t]);
    }
  }

  // Tail rows (nrows % 32) — simple scalar path, first wave of block 0.
  if (blockIdx.x == 0 && w == 0) {
    for (int row = (nTiles << 5) + lane; row < nrows; row += 32) {
      const int2 e = cidx[row];
      const float4 m = mean4[e.x];
      const float4 s = sct4[e.x];
      const size_t ob = (size_t)row * 35u;
      __builtin_nontemporal_store((coords[e.y * 3 + 0] - m.x) * s.x + s.y, &out[ob + 0]);
      __builtin_nontemporal_store((coords[e.y * 3 + 1] - m.y) * s.x + s.z, &out[ob + 1]);
      __builtin_nontemporal_store((coords[e.y * 3 + 2] - m.z) * s.x + s.w, &out[ob + 2]);
      const float* fr = feats + ((size_t)(unsigned)e.y << 5);
      for (int k = 0; k < 32; ++k)
        __builtin_nontemporal_store(fr[k], &out[ob + 3 + k]);
    }
  }
}

// ---------------------------------------------------------------------------
extern "C" void kernel_launch(void* const* d_in, const int* in_sizes, int n_in,
                              void* d_out, int out_size, void* d_ws, size_t ws_size,
                              hipStream_t stream) {
  (void)n_in; (void)out_size; (void)ws_size;
  const int2* cidx = (const int2*)d_in[0];
  const float* feats = (const float*)d_in[1];
  const float* coords = (const float*)d_in[2];
  const int* fullscale = (const int*)d_in[3];
  const int* scale = (const int*)d_in[4];
  const int n = in_sizes[0] / 2;  // SUM_NPOINT

  float* mean4 = (float*)d_ws;                 // 256 * 4 floats
  float* sct4 = mean4 + NUM_CLUSTERS * 4;      // 256 * 4 floats

  stats_kernel<<<NUM_CLUSTERS, 256, 0, stream>>>(cidx, coords, fullscale, scale,
                                                 mean4, sct4, n);

  const int nTiles = n >> 5;
  int grid = (nTiles + WAVES_PER_BLOCK - 1) / WAVES_PER_BLOCK;  // 1 tile per wave
  if (grid < 1) grid = 1;
  out_kernel<<<grid, 256, 0, stream>>>(cidx, feats, coords, (const float4*)mean4,
                                       (const float4*)sct4, (float*)d_out, n);
}